// NET1_QKV9_9861244912091
// MI455X (gfx1250) — compile-verified
//
#include <hip/hip_runtime.h>

typedef float v2f __attribute__((ext_vector_type(2)));
typedef float v4f __attribute__((ext_vector_type(4)));
typedef float v8f __attribute__((ext_vector_type(8)));

namespace {
constexpr int BATCH = 16;
constexpr int CIN   = 256;
constexpr int HW    = 64 * 64;   // 4096
constexpr int MTOT  = 768;       // 3 groups x 256 out channels

constexpr int MB = 128;          // block tile M
constexpr int NB = 128;          // block tile N
constexpr int KB = 16;           // K stage through LDS
constexpr int ASTR = KB + 4;     // 20 floats: float4-aligned rows, conflict-free frag reads
constexpr int BSTR = NB + 16;    // 144 floats: 16B-aligned rows, conflict-free frag reads
constexpr int NSTAGE = CIN / KB; // 16
}

__global__ __launch_bounds__(256, 2)
void qkv_grouped_gemm_wmma(const float* __restrict__ X,
                           const float* __restrict__ Wt,
                           const float* __restrict__ bias,
                           float* __restrict__ Y)
{
    __shared__ float As[MB * ASTR];   // 128 x 16 W tile (m-major)
    __shared__ float Bs[KB * BSTR];   // 16 x 128 X tile (k-major)
    __shared__ float BiasS[MB];

    const int tid  = threadIdx.x;
    const int lane = tid & 31;
    const int wid  = tid >> 5;
    const int waveM = wid >> 2;       // 2 waves along M
    const int waveN = wid & 3;        // 4 waves along N

    const int blockM = blockIdx.y * MB;
    const int blockN = blockIdx.x * NB;
    const int batch  = blockIdx.z;

    const float* Xb = X + (size_t)batch * CIN * HW + blockN;
    const float* Wp = Wt + (size_t)blockM * CIN;
    float* Yp = Y + ((size_t)batch * MTOT + (size_t)blockM) * HW + blockN;

    if (tid < MB) BiasS[tid] = bias[blockM + tid];

    v8f acc[4][2];
#pragma unroll
    for (int mi = 0; mi < 4; ++mi)
#pragma unroll
        for (int ni = 0; ni < 2; ++ni)
            acc[mi][ni] = (v8f)(0.0f);

    // global->LDS staging assignments
    const int ar = tid >> 2;          // 0..63  (A rows; also +64)
    const int ac = (tid & 3) << 2;    // 0,4,8,12
    const int br = tid >> 5;          // 0..7   (B rows; also +8)
    const int bc = (tid & 31) << 2;   // 0..124

    // WMMA fragment lane mapping (ISA 7.12.2, 32-bit operands)
    const int fm = lane & 15;             // m (A) / n (B) within 16
    const int kh = (lane >> 4) << 1;      // K sub-offset: 0 or 2

    for (int st = 0; st < NSTAGE; ++st) {
        const int k0 = st * KB;
        __syncthreads();
        {   // A: 128x16 of W (row stride CIN)
            v4f a0 = *(const v4f*)(Wp + (size_t)ar * CIN + k0 + ac);
            v4f a1 = *(const v4f*)(Wp + (size_t)(ar + 64) * CIN + k0 + ac);
            *(v4f*)&As[ar * ASTR + ac] = a0;
            *(v4f*)&As[(ar + 64) * ASTR + ac] = a1;
        }
        {   // B: 16x128 of X (row stride HW)
            v4f b0 = *(const v4f*)(Xb + (size_t)(k0 + br) * HW + bc);
            v4f b1 = *(const v4f*)(Xb + (size_t)(k0 + br + 8) * HW + bc);
            *(v4f*)&Bs[br * BSTR + bc] = b0;
            *(v4f*)&Bs[(br + 8) * BSTR + bc] = b1;
        }
        if (st + 1 < NSTAGE) {  // prefetch next X stage (global_prefetch_b8)
            __builtin_prefetch(Xb + (size_t)(k0 + KB + br) * HW + bc, 0, 1);
        }
        __syncthreads();

#pragma unroll
        for (int h = 0; h < KB / 4; ++h) {
            const int k = h * 4 + kh;   // per-lane K base (j=0,1 below)
            v2f afrag[4];
#pragma unroll
            for (int mi = 0; mi < 4; ++mi) {
                const int m = waveM * 64 + mi * 16 + fm;
                afrag[mi] = *(const v2f*)&As[m * ASTR + k];   // 8B aligned
            }
            v2f bfrag[2];
#pragma unroll
            for (int ni = 0; ni < 2; ++ni) {
                const int n = waveN * 32 + ni * 16 + fm;
                v2f t;
                t.x = Bs[(k + 0) * BSTR + n];
                t.y = Bs[(k + 1) * BSTR + n];
                bfrag[ni] = t;
            }
#pragma unroll
            for (int mi = 0; mi < 4; ++mi)
#pragma unroll
                for (int ni = 0; ni < 2; ++ni)
                    acc[mi][ni] = __builtin_amdgcn_wmma_f32_16x16x4_f32(
                        false, afrag[mi], false, bfrag[ni],
                        (short)0, acc[mi][ni], false, false);
        }
    }

    // epilogue: bias add + streaming (non-temporal) stores
    const int hsel = lane >> 4;
#pragma unroll
    for (int mi = 0; mi < 4; ++mi) {
#pragma unroll
        for (int ni = 0; ni < 2; ++ni) {
            const int n = waveN * 32 + ni * 16 + fm;
#pragma unroll
            for (int r = 0; r < 8; ++r) {
                const int m = waveM * 64 + mi * 16 + hsel * 8 + r;
                float v = acc[mi][ni][r] + BiasS[m];
                __builtin_nontemporal_store(v, Yp + (size_t)m * HW + n);
            }
        }
    }
}

extern "C" void kernel_launch(void* const* d_in, const int* in_sizes, int n_in,
                              void* d_out, int out_size, void* d_ws, size_t ws_size,
                              hipStream_t stream) {
    const float* x0 = (const float*)d_in[0];   // [16, 256, 64, 64] f32
    const float* W  = (const float*)d_in[1];   // [768, 256] f32
    const float* b  = (const float*)d_in[2];   // [768] f32
    float* y = (float*)d_out;                  // [16, 768, 64, 64] f32

    dim3 grid(HW / NB, MTOT / MB, BATCH);      // (32, 6, 16); x-major => X tiles reused in L2
    dim3 block(256);
    hipLaunchKernelGGL(qkv_grouped_gemm_wmma, grid, block, 0, stream, x0, W, b, y);
}